// CustomGroupedConv2D_70669391888691
// MI455X (gfx1250) — compile-verified
//
#include <hip/hip_runtime.h>

typedef __attribute__((ext_vector_type(2))) float v2f;
typedef __attribute__((ext_vector_type(8))) float v8f;

#define NB 16
#define CIN 256
#define CING 16
#define COUT 512
#define COUTG 32
#define HH 128
#define WW 128
#define HWSZ (HH * WW)

#define TILES_PER_WAVE 8
#define PIXTILES (NB * HH * (WW / 16))           /* 16384 pixel tiles per (g,mtile) */
#define WAVES_PER_GM (PIXTILES / TILES_PER_WAVE) /* 2048 */
#define TOTAL_WAVES (32 * WAVES_PER_GM)          /* 65536 */
#define THREADS 256
#define NBLOCKS (TOTAL_WAVES / (THREADS / 32))   /* 8192 */

__global__ __launch_bounds__(THREADS)
void grouped_conv_wmma_f32(const float* __restrict__ xp,
                           const float* __restrict__ wp,
                           const float* __restrict__ bp,
                           float* __restrict__ yp) {
    const int lane    = threadIdx.x & 31;
    const int lane_lo = lane & 15;   // M for A, N(pixel) for B/C/D
    const int lane_hi = lane >> 4;   // selects K pair / M-half

    const int gw   = (int)((blockIdx.x * blockDim.x + threadIdx.x) >> 5); // global wave id
    const int gm   = gw / WAVES_PER_GM;   // 0..31 : (group, mtile)
    const int widx = gw % WAVES_PER_GM;
    const int g        = gm >> 1;
    const int mbase    = (gm & 1) * 16;
    const int coutbase = g * COUTG + mbase;
    const int xchanbase = g * CING;

    // ---- Preload A fragments (weights): 9 taps x 4 ksteps, reused over 8 tiles ----
    v2f wfrag[36];
#pragma unroll
    for (int t = 0; t < 9; ++t) {
        const int kh = t / 3, kw = t % 3;
#pragma unroll
        for (int s = 0; s < 4; ++s) {
            const int c0 = s * 4 + 2 * lane_hi;
            const int m  = coutbase + lane_lo;
            wfrag[t * 4 + s].x = wp[((m * CING + c0    ) * 3 + kh) * 3 + kw];
            wfrag[t * 4 + s].y = wp[((m * CING + c0 + 1) * 3 + kh) * 3 + kw];
        }
    }
    // ---- Preload bias fragment (matches C/D VGPR layout) ----
    float bias[8];
#pragma unroll
    for (int r = 0; r < 8; ++r)
        bias[r] = bp[coutbase + r + 8 * lane_hi];

#pragma unroll 1
    for (int ti = 0; ti < TILES_PER_WAVE; ++ti) {
        const int pt    = widx * TILES_PER_WAVE + ti;
        const int n_img = pt >> 10;          // / (128 rows * 8 tiles/row)
        const int rem   = pt & 1023;
        const int h     = rem >> 3;
        const int w0    = (rem & 7) << 4;

        v8f acc = {0.f, 0.f, 0.f, 0.f, 0.f, 0.f, 0.f, 0.f};

#pragma unroll
        for (int t = 0; t < 9; ++t) {
            const int kh   = t / 3, kw = t % 3;
            const int hrow = h + kh - 1;
            const int colx = w0 + lane_lo + kw - 1;
            const bool ok  = ((unsigned)hrow < (unsigned)HH) &
                             ((unsigned)colx < (unsigned)WW);
            // clamp so the speculative load is always in-bounds; select 0 for padding
            const int hc = hrow < 0 ? 0 : (hrow > HH - 1 ? HH - 1 : hrow);
            const int wc = colx < 0 ? 0 : (colx > WW - 1 ? WW - 1 : colx);
            const int base = ((n_img * CIN + xchanbase) * HH + hc) * WW + wc;
#pragma unroll
            for (int s = 0; s < 4; ++s) {
                const int c0 = s * 4 + 2 * lane_hi;
                const float v0 = xp[base + c0 * HWSZ];
                const float v1 = xp[base + (c0 + 1) * HWSZ];
                v2f bfrag;
                bfrag.x = ok ? v0 : 0.0f;
                bfrag.y = ok ? v1 : 0.0f;
                acc = __builtin_amdgcn_wmma_f32_16x16x4_f32(
                    /*neg_a=*/false, wfrag[t * 4 + s],
                    /*neg_b=*/false, bfrag,
                    /*c_mod=*/(short)0, acc,
                    /*reuse_a=*/false, /*reuse_b=*/false);
            }
        }

        // ---- Epilogue: bias add + coalesced stores (NCHW) ----
#pragma unroll
        for (int r = 0; r < 8; ++r) {
            const int cout = coutbase + r + 8 * lane_hi;
            const int oidx = ((n_img * COUT + cout) * HH + h) * WW + (w0 + lane_lo);
            yp[oidx] = acc[r] + bias[r];
        }
    }
}

extern "C" void kernel_launch(void* const* d_in, const int* in_sizes, int n_in,
                              void* d_out, int out_size, void* d_ws, size_t ws_size,
                              hipStream_t stream) {
    const float* x  = (const float*)d_in[0];
    const float* Wt = (const float*)d_in[1];
    const float* b  = (const float*)d_in[2];
    float* y        = (float*)d_out;
    grouped_conv_wmma_f32<<<NBLOCKS, THREADS, 0, stream>>>(x, Wt, b, y);
}